// SAM2Teacher_10660108829587
// MI455X (gfx1250) — compile-verified
//
#include <hip/hip_runtime.h>
#include <hip/hip_bf16.h>
#include <math.h>

// Problem constants (match reference)
constexpr int kB = 2, kF = 32, kT = 1024, kD = 1024;
constexpr int kH = 512, kW = 512;
constexpr float kBaseBlend = 1.0f - 0.9f;   // 1 - DECAY
constexpr float kInvTemp   = 1.0f / 0.07f;  // 1 / ATTN_TEMP
constexpr float kCosEps    = 1e-6f;

typedef float v4f __attribute__((ext_vector_type(4)));

// ---------------------------------------------------------------------------
// Kernel 1: coverage per (b,f) = mean(clip(mask,0,1)) over 512x512.
// Streaming reduction, NT loads (data touched exactly once -> bypass L2
// residency). Constant trip count so the scheduler can pipeline loads.
// ---------------------------------------------------------------------------
__global__ __launch_bounds__(256) void sam2_cov_kernel(const float* __restrict__ masks,
                                                       float* __restrict__ cov) {
  const int bf  = blockIdx.x;   // 0..63
  const int tid = threadIdx.x;
  const v4f* p4 = (const v4f*)(masks + (size_t)bf * (size_t)(kH * kW));
  constexpr int kIters = (kH * kW) / 4 / 256;  // 256 float4 per thread
  float s = 0.0f;
#pragma unroll 4
  for (int i = 0; i < kIters; ++i) {
    v4f m = __builtin_nontemporal_load(p4 + tid + i * 256);
#pragma unroll
    for (int k = 0; k < 4; ++k) s += fminf(fmaxf(m[k], 0.0f), 1.0f);
  }
#pragma unroll
  for (int off = 16; off > 0; off >>= 1) s += __shfl_xor(s, off, 32);
  __shared__ float part[8];
  if ((tid & 31) == 0) part[tid >> 5] = s;
  __syncthreads();
  if (tid == 0) {
    float tot = 0.0f;
#pragma unroll
    for (int w = 0; w < 8; ++w) tot += part[w];
    cov[bf] = tot * (1.0f / (float)(kH * kW));
  }
}

// ---------------------------------------------------------------------------
// Kernel 2: per-(b,t)-row scan over F frames.
// Async double-buffered global->LDS streaming (gfx1250 ASYNCcnt path, NT),
// wave32 shuffle reductions for the cosine gate, NT stores of the output.
// One 256-thread block per row; each lane holds 4 floats of memory state.
// ---------------------------------------------------------------------------
__global__ __launch_bounds__(256) void sam2_scan_kernel(const float* __restrict__ toks,
                                                        const float* __restrict__ cov,
                                                        float* __restrict__ out) {
  __shared__ __align__(16) float buf[2][kD];  // 2 x 4KB double buffer
  __shared__ float red[8 * 4];                // 8 waves x {num, |c|^2, |m|^2, pad}

  const int tid  = threadIdx.x;
  const int lane = tid & 31;
  const int wv   = tid >> 5;
  const int b    = blockIdx.x / kT;
  const int t    = blockIdx.x % kT;

  const size_t frameStride = (size_t)kT * kD;                         // elems between frames
  const size_t rowBase     = ((size_t)b * kF * kT + t) * (size_t)kD;  // (b, f=0, t, 0)

  // LDS byte offset of buf (low 32 bits of the generic address of an LDS object)
  const uint32_t ldsBase = (uint32_t)(uintptr_t)(void*)&buf[0][0];
  const uint32_t vOff    = (uint32_t)tid * 16u;  // 16B per lane -> 4KB per block

  // Prefetch frames 0 and 1 into the two buffers (ASYNCcnt = 2).
  {
    const uint64_t g0 = (uint64_t)(uintptr_t)(toks + rowBase);
    const uint64_t g1 = (uint64_t)(uintptr_t)(toks + rowBase + frameStride);
    asm volatile("global_load_async_to_lds_b128 %0, %1, %2 th:TH_LOAD_NT"
                 :: "v"(ldsBase + vOff), "v"(vOff), "s"(g0) : "memory");
    asm volatile("global_load_async_to_lds_b128 %0, %1, %2 th:TH_LOAD_NT"
                 :: "v"(ldsBase + 4096u + vOff), "v"(vOff), "s"(g1) : "memory");
  }

  // ---- Peeled frame 0: memory := frame0, no gate math needed. ----
  asm volatile("s_wait_asynccnt 1" ::: "memory");  // frame 0 landed (frame 1 in flight)
  __syncthreads();                                 // fills visible to all waves
  v4f mem = *(const v4f*)&buf[0][tid * 4];         // ds_load_b128
  __builtin_nontemporal_store(mem, (v4f*)(out + rowBase) + tid);
  __syncthreads();                                 // all reads of buf[0] done -> free
  {
    const uint64_t g2 = (uint64_t)(uintptr_t)(toks + rowBase + 2 * frameStride);
    asm volatile("global_load_async_to_lds_b128 %0, %1, %2 th:TH_LOAD_NT"
                 :: "v"(ldsBase + vOff), "v"(vOff), "s"(g2) : "memory");
  }

  // ---- Frames 1..31 ----
  for (int f = 1; f < kF; ++f) {
    if (f < kF - 1) asm volatile("s_wait_asynccnt 1" ::: "memory");
    else            asm volatile("s_wait_asynccnt 0" ::: "memory");
    __syncthreads();  // all waves' async fills of buf[f&1] visible

    const v4f cur = *(const v4f*)&buf[f & 1][tid * 4];  // ds_load_b128

    // Per-lane partial dots over this lane's 4 elements.
    float pn = cur[0]*mem[0] + cur[1]*mem[1] + cur[2]*mem[2] + cur[3]*mem[3];
    float pc = cur[0]*cur[0] + cur[1]*cur[1] + cur[2]*cur[2] + cur[3]*cur[3];
    float pm = mem[0]*mem[0] + mem[1]*mem[1] + mem[2]*mem[2] + mem[3]*mem[3];
#pragma unroll
    for (int off = 16; off > 0; off >>= 1) {
      pn += __shfl_xor(pn, off, 32);
      pc += __shfl_xor(pc, off, 32);
      pm += __shfl_xor(pm, off, 32);
    }
    if (lane == 0) { red[wv*4+0] = pn; red[wv*4+1] = pc; red[wv*4+2] = pm; }
    __syncthreads();  // partials visible; all reads of buf[f&1] done -> buffer free

    float num = 0.0f, cc = 0.0f, mm = 0.0f;
#pragma unroll
    for (int w = 0; w < 8; ++w) {
      num += red[w*4+0]; cc += red[w*4+1]; mm += red[w*4+2];
    }
    const float sim   = num / (fmaxf(sqrtf(cc), kCosEps) * fmaxf(sqrtf(mm), kCosEps));
    const float gate  = (1.0f / (1.0f + __expf(-sim * kInvTemp)))
                      * (0.5f + 0.5f * cov[b * kF + f]);
    const float blend = fminf(fmaxf(kBaseBlend * (0.5f + gate), 0.0f), 1.0f);
#pragma unroll
    for (int k = 0; k < 4; ++k) mem[k] += blend * (cur[k] - mem[k]);

    // Stream out the new memory state (written once; bypass cache residency).
    __builtin_nontemporal_store(mem, (v4f*)(out + rowBase + (size_t)f * frameStride) + tid);

    // Kick off frame f+2 into the buffer we just freed (buf[f&1]).
    if (f + 2 < kF) {
      const uint64_t g = (uint64_t)(uintptr_t)(toks + rowBase + (size_t)(f + 2) * frameStride);
      asm volatile("global_load_async_to_lds_b128 %0, %1, %2 th:TH_LOAD_NT"
                   :: "v"(ldsBase + (uint32_t)((f & 1) * 4096) + vOff), "v"(vOff), "s"(g)
                   : "memory");
    }
  }
}

// ---------------------------------------------------------------------------
extern "C" void kernel_launch(void* const* d_in, const int* in_sizes, int n_in,
                              void* d_out, int out_size, void* d_ws, size_t ws_size,
                              hipStream_t stream) {
  const float* frame_tokens = (const float*)d_in[0];  // [B,F,T,D] f32
  const float* frame_masks  = (const float*)d_in[1];  // [B,F,H,W] f32
  float* out = (float*)d_out;                         // [B,F,T,D] f32
  float* cov = (float*)d_ws;                          // [B*F] f32 scratch

  sam2_cov_kernel<<<kB * kF, 256, 0, stream>>>(frame_masks, cov);
  sam2_scan_kernel<<<kB * kT, 256, 0, stream>>>(frame_tokens, cov, out);
}